// PointGroup_74380243632410
// MI455X (gfx1250) — compile-verified
//
#include <hip/hip_runtime.h>
#include <math.h>

typedef float v2f __attribute__((ext_vector_type(2)));
typedef float v8f __attribute__((ext_vector_type(8)));

#define CDIM 64
#define SCLS 20
#define WAVES_PER_BLK 4
#define BLK (WAVES_PER_BLK * 32)
#define GRID_BLKS 2048

// LDS strides (floats), padded so WMMA fragment reads are bank-conflict-free
#define WT_STR 68   // transposed weights: wT[c][k]
#define FT_STR 68   // feat tile: ft[row][k]

// workspace float offsets
#define WS_SUM_XY 0
#define WS_SQ_XY 64
#define WS_SUM_Z 128
#define WS_SQ_Z 192
#define WS_SC_XY 256
#define WS_SH_XY 320
#define WS_SC_Z 384
#define WS_SH_Z 448
#define WS_ACC 512
#define WS_TOTAL 520

#define CFENCE() asm volatile("" ::: "memory")
#define WAIT_DS0() asm volatile("s_wait_dscnt 0x0" ::: "memory")
#define WAIT_ASYNC0() asm volatile("s_wait_asynccnt 0x0" ::: "memory")

__device__ __forceinline__ v8f v8zero() {
  v8f z;
#pragma unroll
  for (int i = 0; i < 8; ++i) z[i] = 0.0f;
  return z;
}

// D(16 x 16*NCT) += ft(16x64) @ wT(64 x 16*NCT, stored transposed) via
// v_wmma_f32_16x16x4_f32.
// A frag: lane l -> row l&15,       K = ks*4 + 2*(l>>4) + {0,1}  (ds_load_b64)
// B frag: lane l -> col ct*16+l&15, K = ks*4 + 2*(l>>4) + {0,1}  (ds_load_b64)
template <int NCT>
__device__ __forceinline__ void wmma_gemm(const float* ft, const float* wT,
                                          v8f* acc, int lane) {
  const int r = lane & 15;
  const int kh = lane >> 4;
  const float* arow = ft + r * FT_STR;
#pragma unroll
  for (int ks = 0; ks < 16; ++ks) {
    const int ka = ks * 4 + 2 * kh;
    v2f a = *reinterpret_cast<const v2f*>(arow + ka);
#pragma unroll
    for (int ct = 0; ct < NCT; ++ct) {
      v2f b = *reinterpret_cast<const v2f*>(wT + (ct * 16 + r) * WT_STR + ka);
      acc[ct] = __builtin_amdgcn_wmma_f32_16x16x4_f32(
          false, a, false, b, (short)0, acc[ct], false, false);
    }
  }
}

// stage a 64x64 weight matrix transposed into LDS: wT[c*WT_STR + k] = w[k*64+c]
__device__ __forceinline__ void load_w64T(const float* g, float* lds, int tid) {
  for (int i = tid; i < CDIM * CDIM; i += BLK) {
    int k = i >> 6, c = i & 63;
    lds[c * WT_STR + k] = g[i];
  }
}

// async global -> LDS copy of one 16x64 feat tile (8 x b128 per lane)
__device__ __forceinline__ void load_feat_tile_async(const float* feat, long tileRow,
                                                     int N, float* ft, int lane) {
#pragma unroll
  for (int i = 0; i < 8; ++i) {
    int idx4 = i * 32 + lane;  // 256 float4 = 16 rows x 64 cols
    int row = idx4 >> 4;
    int c4 = idx4 & 15;
    long gr = tileRow + row;
    if (gr >= N) gr = N - 1;
    const float* gsrc = feat + gr * CDIM + c4 * 4;
    unsigned ldsOff = (unsigned)(uintptr_t)(ft + row * FT_STR + c4 * 4);
    asm volatile("global_load_async_to_lds_b128 %0, %1, off"
                 :: "v"(ldsOff), "v"(gsrc)
                 : "memory");
  }
}

// ---------------- pass 0: zero accumulators ----------------
__global__ void pg_zero(float* ws) {
  int t = blockIdx.x * blockDim.x + threadIdx.x;
  if (t < WS_TOTAL) ws[t] = 0.0f;
}

// ---------------- pass 1: GEMM + batch stats ----------------
__global__ void __launch_bounds__(BLK) pg_pass1(const float* __restrict__ feat,
                                                const float* __restrict__ w1xy,
                                                const float* __restrict__ w1z,
                                                float* __restrict__ ws, int N) {
  __shared__ float s_w1xyT[CDIM * WT_STR];
  __shared__ float s_w1zT[CDIM * WT_STR];
  __shared__ float s_ft[WAVES_PER_BLK * 16 * FT_STR];

  const int tid = threadIdx.x;
  const int lane = tid & 31;
  const int wv = tid >> 5;
  load_w64T(w1xy, s_w1xyT, tid);
  load_w64T(w1z, s_w1zT, tid);
  __syncthreads();

  float* ft = s_ft + wv * 16 * FT_STR;
  const int r = lane & 15;
  const int kh = lane >> 4;

  float sxy[4] = {0, 0, 0, 0}, qxy[4] = {0, 0, 0, 0};
  float sz[4] = {0, 0, 0, 0}, qz[4] = {0, 0, 0, 0};

  const int tiles = (N + 15) >> 4;
  const int wgid = blockIdx.x * WAVES_PER_BLK + wv;
  const int wstride = gridDim.x * WAVES_PER_BLK;

  for (int t = wgid; t < tiles; t += wstride) {
    WAIT_DS0();  // previous tile's LDS reads done before async engine overwrites
    load_feat_tile_async(feat, (long)t * 16, N, ft, lane);
    WAIT_ASYNC0();  // tile resident in LDS
    v8f axy[4], az[4];
#pragma unroll
    for (int i = 0; i < 4; ++i) { axy[i] = v8zero(); az[i] = v8zero(); }
    wmma_gemm<4>(ft, s_w1xyT, axy, lane);
    wmma_gemm<4>(ft, s_w1zT, az, lane);
#pragma unroll
    for (int ct = 0; ct < 4; ++ct)
#pragma unroll
      for (int v = 0; v < 8; ++v) {
        float h = axy[ct][v];  sxy[ct] += h;  qxy[ct] += h * h;
        float hz = az[ct][v];  sz[ct] += hz;  qz[ct] += hz * hz;
      }
  }
  // lanes l and l+16 hold the same channel set: combine, then one atomic burst
#pragma unroll
  for (int ct = 0; ct < 4; ++ct) {
    sxy[ct] += __shfl_xor(sxy[ct], 16, 32);
    qxy[ct] += __shfl_xor(qxy[ct], 16, 32);
    sz[ct]  += __shfl_xor(sz[ct], 16, 32);
    qz[ct]  += __shfl_xor(qz[ct], 16, 32);
  }
  if (kh == 0) {
#pragma unroll
    for (int ct = 0; ct < 4; ++ct) {
      int c = ct * 16 + r;
      atomicAdd(ws + WS_SUM_XY + c, sxy[ct]);
      atomicAdd(ws + WS_SQ_XY + c, qxy[ct]);
      atomicAdd(ws + WS_SUM_Z + c, sz[ct]);
      atomicAdd(ws + WS_SQ_Z + c, qz[ct]);
    }
  }
}

// ---------------- pass 2: fold stats into scale/shift ----------------
__global__ void pg_stats(float* ws, const float* gxy, const float* bexy,
                         const float* gz, const float* bez, int N) {
  int t = threadIdx.x;
  float invN = 1.0f / (float)N;
  if (t < 64) {
    float mu = ws[WS_SUM_XY + t] * invN;
    float var = ws[WS_SQ_XY + t] * invN - mu * mu;
    float sc = gxy[t] * rsqrtf(var + 1e-3f);
    ws[WS_SC_XY + t] = sc;
    ws[WS_SH_XY + t] = bexy[t] - mu * sc;
  } else if (t < 128) {
    int c = t - 64;
    float mu = ws[WS_SUM_Z + c] * invN;
    float var = ws[WS_SQ_Z + c] * invN - mu * mu;
    float sc = gz[c] * rsqrtf(var + 1e-3f);
    ws[WS_SC_Z + c] = sc;
    ws[WS_SH_Z + c] = bez[c] - mu * sc;
  }
}

// ---------------- pass 3: recompute GEMMs, heads, losses ----------------
__global__ void __launch_bounds__(BLK) pg_pass3(
    const float* __restrict__ feat, const float* __restrict__ coord,
    const int* __restrict__ segment, const int* __restrict__ instance,
    const float* __restrict__ icent,
    const float* __restrict__ w1xy, const float* __restrict__ w1z,
    const float* __restrict__ w2xy, const float* __restrict__ b2xy,
    const float* __restrict__ w2z, const float* __restrict__ b2z,
    const float* __restrict__ wseg, const float* __restrict__ bseg,
    float* __restrict__ ws, int N) {
  __shared__ float s_w1xyT[CDIM * WT_STR];
  __shared__ float s_w1zT[CDIM * WT_STR];
  __shared__ float s_wsegT[32 * WT_STR];
  __shared__ float s_ft[WAVES_PER_BLK * 16 * FT_STR];
  __shared__ float s_scxy[64], s_shxy[64], s_scz[64], s_shz[64];
  __shared__ float s_w2xy[128], s_w2z[64], s_bseg[32];

  const int tid = threadIdx.x;
  const int lane = tid & 31;
  const int wv = tid >> 5;
  load_w64T(w1xy, s_w1xyT, tid);
  load_w64T(w1z, s_w1zT, tid);
  for (int i = tid; i < CDIM * 32; i += BLK) {
    int k = i >> 5, c = i & 31;  // wsegT[c][k], zero-pad classes 20..31
    s_wsegT[c * WT_STR + k] = (c < SCLS) ? wseg[k * SCLS + c] : 0.0f;
  }
  for (int i = tid; i < 64; i += BLK) {
    s_scxy[i] = ws[WS_SC_XY + i];
    s_shxy[i] = ws[WS_SH_XY + i];
    s_scz[i] = ws[WS_SC_Z + i];
    s_shz[i] = ws[WS_SH_Z + i];
    s_w2z[i] = w2z[i];
  }
  for (int i = tid; i < 128; i += BLK) s_w2xy[i] = w2xy[i];
  for (int i = tid; i < 32; i += BLK) s_bseg[i] = (i < SCLS) ? bseg[i] : 0.0f;
  __syncthreads();

  float* ft = s_ft + wv * 16 * FT_STR;
  const int r = lane & 15;
  const int kh = lane >> 4;
  const float b2x = b2xy[0], b2y = b2xy[1], b2zz = b2z[0];

  float aNll = 0, aValid = 0, aMask = 0, aL1xy = 0, aCosxy = 0, aL1z = 0, aCosz = 0;

  const int tiles = (N + 15) >> 4;
  const int wgid = blockIdx.x * WAVES_PER_BLK + wv;
  const int wstride = gridDim.x * WAVES_PER_BLK;

  for (int t = wgid; t < tiles; t += wstride) {
    WAIT_DS0();
    load_feat_tile_async(feat, (long)t * 16, N, ft, lane);
    WAIT_ASYNC0();
    v8f axy[4], az[4], asg[2];
#pragma unroll
    for (int i = 0; i < 4; ++i) { axy[i] = v8zero(); az[i] = v8zero(); }
    asg[0] = v8zero(); asg[1] = v8zero();
    wmma_gemm<4>(ft, s_w1xyT, axy, lane);
    wmma_gemm<4>(ft, s_w1zT, az, lane);
    wmma_gemm<2>(ft, s_wsegT, asg, lane);
    CFENCE();

    // ---- xy head: BN + ReLU into LDS tile (reuse ft), then 64-dot with W2_xy ----
#pragma unroll
    for (int ct = 0; ct < 4; ++ct)
#pragma unroll
      for (int v = 0; v < 8; ++v) {
        int c = ct * 16 + r, m = v + 8 * kh;
        float h = fmaf(axy[ct][v], s_scxy[c], s_shxy[c]);
        ft[m * FT_STR + c] = fmaxf(h, 0.0f);
      }
    CFENCE();
    float px = 0, py = 0;
#pragma unroll
    for (int i = 0; i < 32; ++i) {
      int c = kh * 32 + i;
      float h = ft[r * FT_STR + c];
      px = fmaf(h, s_w2xy[c * 2 + 0], px);
      py = fmaf(h, s_w2xy[c * 2 + 1], py);
    }
    px += __shfl_xor(px, 16, 32); py += __shfl_xor(py, 16, 32);
    px += b2x; py += b2y;
    CFENCE();

    // ---- z head ----
#pragma unroll
    for (int ct = 0; ct < 4; ++ct)
#pragma unroll
      for (int v = 0; v < 8; ++v) {
        int c = ct * 16 + r, m = v + 8 * kh;
        float h = fmaf(az[ct][v], s_scz[c], s_shz[c]);
        ft[m * FT_STR + c] = fmaxf(h, 0.0f);
      }
    CFENCE();
    float pz = 0;
#pragma unroll
    for (int i = 0; i < 32; ++i) {
      int c = kh * 32 + i;
      pz = fmaf(ft[r * FT_STR + c], s_w2z[c], pz);
    }
    pz += __shfl_xor(pz, 16, 32);
    pz += b2zz;
    CFENCE();

    // ---- seg logits -> log-softmax CE (20 classes, 10 per half-wave) ----
#pragma unroll
    for (int ct = 0; ct < 2; ++ct)
#pragma unroll
      for (int v = 0; v < 8; ++v) {
        int c = ct * 16 + r, m = v + 8 * kh;
        ft[m * FT_STR + c] = asg[ct][v] + s_bseg[c];
      }
    CFENCE();
    float mx = -3.4e38f;
#pragma unroll
    for (int i = 0; i < 10; ++i) mx = fmaxf(mx, ft[r * FT_STR + kh * 10 + i]);
    mx = fmaxf(mx, __shfl_xor(mx, 16, 32));
    float se = 0;
#pragma unroll
    for (int i = 0; i < 10; ++i) se += __expf(ft[r * FT_STR + kh * 10 + i] - mx);
    se += __shfl_xor(se, 16, 32);

    long gidx = (long)t * 16 + r;
    bool inb = gidx < (long)N;
    long gcl = inb ? gidx : (long)(N - 1);
    int sg = segment[gcl];
    bool valid = (sg >= 0);
    int tg = valid ? sg : 0;
    float nll = -((ft[r * FT_STR + tg] - mx) - __logf(se));

    if (kh == 0 && inb) {
      if (valid) { aNll += nll; aValid += 1.0f; }
      int inst = instance[gidx];
      float msk = (inst >= 0) ? 1.0f : 0.0f;
      aMask += msk;
      float gx = icent[gidx * 3 + 0] - coord[gidx * 3 + 0];
      float gy = icent[gidx * 3 + 1] - coord[gidx * 3 + 1];
      float gz = icent[gidx * 3 + 2] - coord[gidx * 3 + 2];
      aL1xy += (fabsf(gx - px) + fabsf(gy - py)) * msk;
      float gn = sqrtf(gx * gx + gy * gy) + 1e-8f;
      float pn = sqrtf(px * px + py * py) + 1e-8f;
      aCosxy += -((gx * px + gy * py) / (gn * pn)) * msk;
      aL1z += fabsf(gz - pz) * msk;
      aCosz += -((gz / (fabsf(gz) + 1e-8f)) * (pz / (fabsf(pz) + 1e-8f))) * msk;
    }
  }

  float vals[7] = {aNll, aValid, aMask, aL1xy, aCosxy, aL1z, aCosz};
#pragma unroll
  for (int m = 1; m < 32; m <<= 1)
#pragma unroll
    for (int i = 0; i < 7; ++i) vals[i] += __shfl_xor(vals[i], m, 32);
  if (lane == 0)
#pragma unroll
    for (int i = 0; i < 7; ++i) atomicAdd(ws + WS_ACC + i, vals[i]);
}

// ---------------- pass 4: combine ----------------
__global__ void pg_final(const float* __restrict__ ws, float* __restrict__ out) {
  if (threadIdx.x == 0 && blockIdx.x == 0) {
    float nll = ws[WS_ACC + 0], vf = ws[WS_ACC + 1], msk = ws[WS_ACC + 2];
    float l1xy = ws[WS_ACC + 3], cxy = ws[WS_ACC + 4];
    float l1z = ws[WS_ACC + 5], cz = ws[WS_ACC + 6];
    float segloss = nll / vf;
    float denom = msk + 1e-8f;
    float L1xy = l1xy / denom, Cxy = cxy / denom;
    float L1z = l1z / denom, Cz = cz / denom;
    float loss = segloss + L1xy + L1xy + 0.5f * (L1z + Cz);
    out[0] = loss; out[1] = segloss; out[2] = L1xy;
    out[3] = Cxy;  out[4] = L1z;     out[5] = Cz;
  }
}

extern "C" void kernel_launch(void* const* d_in, const int* in_sizes, int n_in,
                              void* d_out, int out_size, void* d_ws, size_t ws_size,
                              hipStream_t stream) {
  const float* feat = (const float*)d_in[0];
  const float* coord = (const float*)d_in[1];
  const int* segment = (const int*)d_in[2];
  const int* instance = (const int*)d_in[3];
  const float* icent = (const float*)d_in[4];
  const float* w1xy = (const float*)d_in[5];
  // d_in[6] = b1_xy: cancels under batch-stat BN
  const float* gxy = (const float*)d_in[7];
  const float* bexy = (const float*)d_in[8];
  const float* w2xy = (const float*)d_in[9];
  const float* b2xy = (const float*)d_in[10];
  const float* w1z = (const float*)d_in[11];
  // d_in[12] = b1_z: cancels
  const float* gz = (const float*)d_in[13];
  const float* bez = (const float*)d_in[14];
  const float* w2z = (const float*)d_in[15];
  const float* b2z = (const float*)d_in[16];
  const float* wseg = (const float*)d_in[17];
  const float* bseg = (const float*)d_in[18];
  float* out = (float*)d_out;
  float* wsf = (float*)d_ws;
  const int N = in_sizes[0] / CDIM;

  pg_zero<<<1, 1024, 0, stream>>>(wsf);
  pg_pass1<<<GRID_BLKS, BLK, 0, stream>>>(feat, w1xy, w1z, wsf, N);
  pg_stats<<<1, 128, 0, stream>>>(wsf, gxy, bexy, gz, bez, N);
  pg_pass3<<<GRID_BLKS, BLK, 0, stream>>>(feat, coord, segment, instance, icent,
                                          w1xy, w1z, w2xy, b2xy, w2z, b2z,
                                          wseg, bseg, wsf, N);
  pg_final<<<1, 64, 0, stream>>>(wsf, out);
}